// SystemChargeNeutralize_5918464934539
// MI455X (gfx1250) — compile-verified
//
#include <hip/hip_runtime.h>

typedef __attribute__((ext_vector_type(2))) float v2f;
typedef __attribute__((ext_vector_type(4))) float v4f;
typedef __attribute__((ext_vector_type(8))) float v8f;

#define N_ATOMS 2000000
#define N_MOL   50000
#define D_FEAT  128

// ---------------------------------------------------------------- zero scratch
__global__ void zero_ws_kernel(float* __restrict__ ws, int n) {
    int i = blockIdx.x * blockDim.x + threadIdx.x;
    if (i < n) ws[i] = 0.0f;
}

// ---------------------------------------------------------------- q = p1 @ W + b  (WMMA f32 16x16x4)
// One wave per 16-atom tile. A-matrix (16x4 f32): lanes 0-15 carry K=0,1 in
// VGPR0/1; lanes 16-31 carry K=2,3 (ISA 32-bit A layout). B-matrix (4x16)
// broadcasts W[k..k+3] across all 16 columns, which under the B layout means
// every lane's 2 B-VGPRs hold the same (w[k+2*khalf], w[k+2*khalf+1]) pair —
// identical per-lane indexing to A, so a b64 load of W builds it directly.
// D's 16 columns are then all equal to the per-atom partial dot products.
__global__ __launch_bounds__(256)
void matvec_wmma_kernel(const float* __restrict__ p1,
                        const float* __restrict__ W,
                        const float* __restrict__ bias_p,
                        float* __restrict__ q) {
    const int  lane  = threadIdx.x & 31;
    const int  wave  = threadIdx.x >> 5;
    const long tile  = (long)blockIdx.x * 8 + wave;   // 16 atoms per tile
    const long base  = tile * 16;
    const int  row   = lane & 15;                     // atom within tile (M)
    const int  khalf = lane >> 4;                     // 0: K=0,1  1: K=2,3

    const float* prow = p1 + (base + row) * (long)D_FEAT + (khalf << 1);
    const float* wp   = W + (khalf << 1);

    v8f acc = {0.f, 0.f, 0.f, 0.f, 0.f, 0.f, 0.f, 0.f};
#pragma unroll 4
    for (int k0 = 0; k0 < D_FEAT; k0 += 4) {
        v2f a  = *(const v2f*)(prow + k0);   // global_load_b64 (A tile slice)
        v2f bm = *(const v2f*)(wp + k0);     // W chunk, L0/WGP$-resident
        acc = __builtin_amdgcn_wmma_f32_16x16x4_f32(
            /*neg_a=*/false, a, /*neg_b=*/false, bm,
            /*c_mod=*/(short)0, acc, /*reuse_a=*/false, /*reuse_b=*/false);
    }

    const float bias = bias_p[0];
    // D layout: acc[r] on lanes 0-15 = q[base+r]; on lanes 16-31 = q[base+8+r].
    // Columns are replicated, so lane 0 / lane 16 each emit 8 contiguous floats.
    if (row == 0) {
        float* qp = q + base + (khalf << 3);
        v4f lo = { acc[0] + bias, acc[1] + bias, acc[2] + bias, acc[3] + bias };
        v4f hi = { acc[4] + bias, acc[5] + bias, acc[6] + bias, acc[7] + bias };
        *(v4f*)(qp)     = lo;                // global_store_b128
        *(v4f*)(qp + 4) = hi;
    }
}

// ---------------------------------------------------------------- segment sums
// atom_batch is sorted, so within a wave equal ids are contiguous: a segmented
// suffix scan (id-equality is a valid segment test) leaves the full run sum on
// the run's leader lane -> ~1 atomic per molecule-run instead of per atom.
__global__ __launch_bounds__(256)
void segreduce_kernel(const float* __restrict__ q,
                      const int*   __restrict__ ab,
                      float* __restrict__ qsum,
                      float* __restrict__ cnt) {
    int  i     = blockIdx.x * blockDim.x + threadIdx.x;
    int  lane  = threadIdx.x & 31;
    bool valid = (i < N_ATOMS);
    float v = valid ? q[i]  : 0.0f;
    float c = valid ? 1.0f  : 0.0f;
    int   id = valid ? ab[i] : -1;

    #pragma unroll
    for (int off = 1; off < 32; off <<= 1) {
        float nv  = __shfl_down(v,  off, 32);
        float nc  = __shfl_down(c,  off, 32);
        int   nid = __shfl_down(id, off, 32);
        if (lane + off < 32 && nid == id) { v += nv; c += nc; }
    }
    int  pid    = __shfl_up(id, 1, 32);
    bool leader = valid && (lane == 0 || pid != id);
    if (leader) {
        unsafeAtomicAdd(&qsum[id], v);   // global_atomic_add_f32
        unsafeAtomicAdd(&cnt[id],  c);
    }
}

// ---------------------------------------------------------------- per-mol mean
__global__ void pcharge_kernel(const float* __restrict__ qsum,
                               const float* __restrict__ cnt,
                               float* __restrict__ pc) {
    int m = blockIdx.x * blockDim.x + threadIdx.x;
    if (m < N_MOL) pc[m] = qsum[m] / fmaxf(cnt[m], 1.0f);
}

// ---------------------------------------------------------------- neutralize
__global__ void finalize_kernel(const int* __restrict__ ab,
                                const float* __restrict__ pc,
                                float* __restrict__ out) {
    int i = blockIdx.x * blockDim.x + threadIdx.x;
    if (i < N_ATOMS) out[i] = out[i] - pc[ab[i]];
}

// ---------------------------------------------------------------- launch
extern "C" void kernel_launch(void* const* d_in, const int* in_sizes, int n_in,
                              void* d_out, int out_size, void* d_ws, size_t ws_size,
                              hipStream_t stream) {
    const int*   ab = (const int*)  d_in[0];   // atom_batch (int32, sorted)
    const float* p1 = (const float*)d_in[1];   // [N_ATOMS, 128]
    const float* W  = (const float*)d_in[2];   // [128]
    const float* b  = (const float*)d_in[3];   // [1]
    float* out = (float*)d_out;                // q, then neutralized in place
    float* ws  = (float*)d_ws;
    float* qsum = ws;                          // [N_MOL]
    float* cnt  = ws +     N_MOL;              // [N_MOL]
    float* pc   = ws + 2 * N_MOL;              // [N_MOL]   (600 KB total)

    zero_ws_kernel   <<<(2 * N_MOL + 255) / 256, 256, 0, stream>>>(ws, 2 * N_MOL);
    matvec_wmma_kernel<<<N_ATOMS / 128,          256, 0, stream>>>(p1, W, b, out);
    segreduce_kernel <<<(N_ATOMS + 255) / 256,   256, 0, stream>>>(out, ab, qsum, cnt);
    pcharge_kernel   <<<(N_MOL  + 255) / 256,    256, 0, stream>>>(qsum, cnt, pc);
    finalize_kernel  <<<(N_ATOMS + 255) / 256,   256, 0, stream>>>(ab, pc, out);
}